// features_38319698215238
// MI455X (gfx1250) — compile-verified
//
#include <hip/hip_runtime.h>
#include <stdint.h>

// Problem constants (from reference)
#define BB 256
#define LL 32768

// Tiling
#define TILE 4096                 // output columns per block
#define HALO 100                  // >= 99 pad for k=199 window, multiple of 4 for b128 alignment
#define NTOT (TILE + 2 * HALO)    // 4296 staged elements
#define NV4  (NTOT / 4)           // 1074 float4 async loads
#define NTHREADS 256              // 8 wave32s
#define CH 17                     // ceil(NTOT / NTHREADS) elements scanned per thread
#define NPAD (CH * NTHREADS)      // 4352 padded scan length
#define TILES_PER_ROW (LL / TILE) // 8

typedef float v4f __attribute__((ext_vector_type(4)));

__global__ __launch_bounds__(NTHREADS) void features_pool_kernel(
    const float* __restrict__ x, float* __restrict__ out) {
  __shared__ __align__(16) float stage[NTOT];     // raw x tile (+halo)
  __shared__ float Pa[NPAD + 1];                  // prefix of a = log(max(1,1+x))
  __shared__ float Pb[NPAD + 1];                  // prefix of b = log(max(1,1-x))
  __shared__ float totA[NTHREADS];
  __shared__ float totB[NTHREADS];

  const int t    = threadIdx.x;
  const int row  = blockIdx.x / TILES_PER_ROW;
  const int tile = blockIdx.x % TILES_PER_ROW;
  const int t0   = tile * TILE;

  const float* __restrict__ xrow = x + (size_t)row * LL;

  // ---- Stage 1: CDNA5 async global->LDS copy of x[t0-HALO .. t0+TILE+HALO) ----
  // Addresses are clamped into the row (16B aligned); out-of-range entries are
  // garbage in LDS but are masked at the a/b computation (zero padding there).
#pragma unroll
  for (int k = 0; k < (NV4 + NTHREADS - 1) / NTHREADS; ++k) {
    int idx = t + k * NTHREADS;
    if (idx < NV4) {
      int gc = t0 - HALO + idx * 4;
      gc = gc < 0 ? 0 : (gc > (LL - 4) ? (LL - 4) : gc);
      // low 32 bits of the flat LDS-aperture address == LDS byte offset
      uint32_t lds_addr = (uint32_t)(uintptr_t)(&stage[idx * 4]);
      uint64_t gaddr    = (uint64_t)(uintptr_t)(xrow + gc);
      asm volatile("global_load_async_to_lds_b128 %0, %1, off"
                   :: "v"(lds_addr), "v"(gaddr)
                   : "memory");
    }
  }
  asm volatile("s_wait_asynccnt 0" ::: "memory");  // ASYNCcnt -> 0
  __syncthreads();

  // ---- Stage 2: per-thread sequential scan of a,b over its 17-element chunk ----
  const int j0 = t * CH;
  float runA = 0.f, runB = 0.f;
#pragma unroll
  for (int jj = 0; jj < CH; ++jj) {
    int j = j0 + jj;
    float a = 0.f, b = 0.f;
    if (j < NTOT) {
      int gc = t0 - HALO + j;
      if (gc >= 0 && gc < LL) {
        float v = stage[j];
        a = __logf(fmaxf(1.0f, 1.0f + v));
        b = __logf(fmaxf(1.0f, 1.0f - v));
      }
    }
    runA += a;
    runB += b;
    Pa[j + 1] = runA;   // local inclusive prefix (offset fixed up below)
    Pb[j + 1] = runB;
  }
  totA[t] = runA;
  totB[t] = runB;
  if (t == 0) { Pa[0] = 0.f; Pb[0] = 0.f; }
  __syncthreads();

  // ---- Hillis-Steele inclusive scan over the 256 chunk totals ----
#pragma unroll
  for (int s = 1; s < NTHREADS; s <<= 1) {
    float aL = (t >= s) ? totA[t - s] : 0.f;
    float bL = (t >= s) ? totB[t - s] : 0.f;
    __syncthreads();
    totA[t] += aL;
    totB[t] += bL;
    __syncthreads();
  }
  const float offA = totA[t] - runA;  // exclusive offset for this chunk
  const float offB = totB[t] - runB;

  // Fix up own chunk (disjoint regions -> no barrier needed before the adds)
#pragma unroll
  for (int jj = 1; jj <= CH; ++jj) {
    Pa[j0 + jj] += offA;
    Pb[j0 + jj] += offB;
  }
  __syncthreads();

  // ---- Stage 3: windows from prefix sums, 4-channel 128-bit NT stores ----
  // a0[i] = (P[li+2]   - P[li-1] ) / 3     (k=3,  p=1)
  // a5[i] = (P[li+100] - P[li-99]) / 199   (k=199,p=99), li = i - t0 + HALO
  const float inv3   = (1.0f / 3.0f);
  const float inv199 = (1.0f / 199.0f);
  const size_t obase = (size_t)row * 4 * LL;

#pragma unroll
  for (int k = 0; k < TILE / (NTHREADS * 4); ++k) {   // 4 iterations
    const int v  = (k * NTHREADS + t) * 4;            // tile-local col, x4 vectorized
    const int li = v + HALO;
    v4f o0, o1, o2, o3;
#pragma unroll
    for (int j = 0; j < 4; ++j) {
      int m = li + j;
      float a0 = (Pa[m + 2]   - Pa[m - 1])  * inv3;
      float a5 = (Pa[m + 100] - Pa[m - 99]) * inv199;
      float b0 = (Pb[m + 2]   - Pb[m - 1])  * inv3;
      float b5 = (Pb[m + 100] - Pb[m - 99]) * inv199;
      o0[j] = a0 - a5;
      o1[j] = a0;
      o2[j] = b0 - b5;
      o3[j] = b0;
    }
    const size_t col = (size_t)(t0 + v);
    __builtin_nontemporal_store(o0, (v4f*)(out + obase + 0 * (size_t)LL + col));
    __builtin_nontemporal_store(o1, (v4f*)(out + obase + 1 * (size_t)LL + col));
    __builtin_nontemporal_store(o2, (v4f*)(out + obase + 2 * (size_t)LL + col));
    __builtin_nontemporal_store(o3, (v4f*)(out + obase + 3 * (size_t)LL + col));
  }
}

extern "C" void kernel_launch(void* const* d_in, const int* in_sizes, int n_in,
                              void* d_out, int out_size, void* d_ws, size_t ws_size,
                              hipStream_t stream) {
  const float* x = (const float*)d_in[0];
  float* out     = (float*)d_out;
  dim3 grid(BB * TILES_PER_ROW);   // 2048 blocks
  dim3 block(NTHREADS);            // 8 wave32s per block
  features_pool_kernel<<<grid, block, 0, stream>>>(x, out);
}